// HOG_48756468744884
// MI455X (gfx1250) — compile-verified
//
#include <hip/hip_runtime.h>

// HOG features for MI455X (gfx1250).
// Memory-bound (201 MB in / 2 MB out -> ~8.7us floor at 23.3 TB/s); no GEMM
// structure, so WMMA would only add work. The CDNA5 feature that fits the
// roofline is the Tensor Data Mover: double-buffered async 3D tile loads
// (global -> LDS) tracked with TENSORcnt, overlapping the HBM stream with the
// per-pixel gradient/binning VALU work.

typedef unsigned int       u32;
typedef unsigned long long u64;
typedef u32 u32x4 __attribute__((ext_vector_type(4)));
typedef int i32x4 __attribute__((ext_vector_type(4)));
typedef int i32x8 __attribute__((ext_vector_type(8)));

#define HOG_H 512
#define HOG_W 512
#define HOG_CH 3
#define CELL 16
#define NCC 32            // 512/16 cell cols
#define BLK_CELLS 4       // cells per pipeline step
#define BLK_COLS 64       // 4*16
#define NBLK 8            // 512/64
#define TILE_H 18         // 16 rows + 2 halo
#define TILE_W 66         // 64 cols + 2 halo
#define TILE_ELEMS (HOG_CH * TILE_H * TILE_W)   // 3564 floats

// Issue one TDM 3D tile load: tile [3][<=18][<=66] f32 from the image of
// batch b, strip row cr*16-1, cols blk*64-1 into LDS at lds_byte.
// Top/left edges: clamp origin, shrink tile dims (compute offsets the
// indexing identically). Right/bottom: tensor_dim = remaining extent so the
// TDM zero-fills OOB reads.
__device__ __forceinline__ void tdm_issue_tile(const float* in, u32 lds_byte,
                                               int b, int cr, int blk) {
  const int  col_lo = (blk == 0) ? 0 : (blk * BLK_COLS - 1);
  const int  row_lo = (cr == 0) ? 0 : (cr * CELL - 1);
  const u32  tile_w = (blk == 0) ? (TILE_W - 1) : TILE_W;
  const u32  tile_h = (cr == 0) ? (TILE_H - 1) : TILE_H;
  const u32  td0 = (u32)(HOG_W - col_lo);         // remaining cols
  const u32  td1 = (u32)(HOG_H - row_lo);         // remaining rows
  const u64  gaddr = (u64)(size_t)in +
      4ull * ((u64)b * (u64)(HOG_CH * HOG_H * HOG_W) +
              (u64)row_lo * (u64)HOG_W + (u64)col_lo);

  // D# group 0: count=1 | lds_addr | global_addr[56:0] | type=2
  u32x4 g0 = { 1u,
               lds_byte,
               (u32)gaddr,
               (u32)((gaddr >> 32) & 0x1FFFFFFull) | (2u << 30) };
  // D# group 1: data_size=2 (4B); tensor_dim0/1; tile_dim0/1/2; strides.
  i32x8 g1 = { (int)0x00020000u,                               // data_size=2
               (int)((td0 & 0xFFFFu) << 16),                   // dim0[15:0] @63:48
               (int)(((td0 >> 16) & 0xFFFFu) | ((td1 & 0xFFFFu) << 16)),
               (int)(((td1 >> 16) & 0xFFFFu) | (tile_w << 16)),// tile_dim0 @127:112
               (int)(tile_h | (3u << 16)),                     // tile_dim1, tile_dim2
               (int)HOG_W,                                     // dim0_stride lo32
               0,                                              // dim0_stride hi16 | dim1_stride[15:0]
               (int)((HOG_H * HOG_W) >> 16) };                 // dim1_stride[47:16]
  i32x4 g2 = { HOG_CH, 0, 0, 0 };                              // tensor_dim2=3
  i32x4 g3 = { 0, 0, 0, 0 };
  // clang-23 prod-lane arity: 6 args with an extra int32x8 slot before cpol
  // (beyond the 4 architectural D# groups); zero-filled per the probe.
  i32x8 g4 = { 0, 0, 0, 0, 0, 0, 0, 0 };
  __builtin_amdgcn_tensor_load_to_lds(g0, g1, g2, g3, g4, 0);
}

__global__ __launch_bounds__(256)
void hog_cdna5_kernel(const float* __restrict__ in, float* __restrict__ out) {
  __shared__ float tile[2][TILE_ELEMS];   // 28512 B double buffer
  __shared__ float wavehist[8 * 8];       // per-wave 8-bin partials

  const int tid  = threadIdx.x;
  const int wid  = tid >> 5;
  const int lane = tid & 31;
  const int cr   = blockIdx.x;            // cell row 0..31
  const int b    = blockIdx.y;            // batch 0..63

  // scalar (SGPR) wave id so the TDM issue sits behind a scalar branch
  const u32 wid_s = __builtin_amdgcn_readfirstlane((u32)wid);

  const int  drow   = (cr == 0) ? 1 : 0;
  const int  th_    = TILE_H - drow;      // tile rows actually written
  const int  half   = wid & 1;            // top/bottom half of cell
  const int  cell_w = wid >> 1;           // cell within 4-cell block
  const int  r_loc  = (half << 3) + (lane >> 2);     // 0..15 row in strip
  const int  r      = (cr << 4) + r_loc;             // global row
  const int  j      = r_loc + 1;                     // tile row (halo at 0)
  const bool row_ok = (r > 0) & (r < HOG_H - 1);
  const int  ibase  = (cell_w << 4) + ((lane & 3) << 2) + 1; // tile col of px0

  if (wid_s == 0) {
    tdm_issue_tile(in, (u32)(size_t)&tile[0][0], b, cr, 0);
  }

  for (int blk = 0; blk < NBLK; ++blk) {
    if (wid_s == 0) {
      if (blk + 1 < NBLK) {
        tdm_issue_tile(in, (u32)(size_t)&tile[(blk + 1) & 1][0], b, cr, blk + 1);
        __builtin_amdgcn_s_wait_tensorcnt(1);   // tile for `blk` resident
      } else {
        __builtin_amdgcn_s_wait_tensorcnt(0);
      }
    }
    __syncthreads();                            // publish tile[blk&1]

    const int    dcol = (blk == 0) ? 1 : 0;
    const int    tw   = TILE_W - dcol;          // LDS row stride this block
    const int    chs  = th_ * tw;               // LDS channel stride
    const float* t    = &tile[blk & 1][0];
    // index(ch,j,i) = ch*chs + (j-drow)*tw + (i-dcol)
    const int roff = j * tw - drow * tw - dcol;

    float acc[8];
#pragma unroll
    for (int o = 0; o < 8; ++o) acc[o] = 0.f;

#pragma unroll
    for (int k = 0; k < 4; ++k) {
      const int  i      = ibase + k;
      const int  c      = (blk << 6) + i - 1;   // global col
      const bool col_ok = (c > 0) & (c < HOG_W - 1);

      // channel with max squared gradient magnitude (first-max on ties)
      float bm = -1.f, gr_s = 0.f, gc_s = 0.f;
#pragma unroll
      for (int ch = 0; ch < HOG_CH; ++ch) {
        const int base = ch * chs + roff + i;
        const float up = t[base - tw];
        const float dn = t[base + tw];
        const float lf = t[base - 1];
        const float rt = t[base + 1];
        const float gr = row_ok ? (dn - up) : 0.f;  // borders -> 0 (skimage)
        const float gc = col_ok ? (rt - lf) : 0.f;
        const float m2 = gr * gr + gc * gc;
        if (m2 > bm) { bm = m2; gr_s = gr; gc_s = gc; }
      }
      const float magn = sqrtf(bm);

      // bin = floor((deg(atan2(gr,gc)) mod 180) / 22.5) via 7 half-plane
      // tests on the canonicalized vector (y>=0; y==0&&x<0 -> negate both,
      // matching the 180 -> 0 wrap).
      float x = gc_s, y = gr_s;
      if (y < 0.f || (y == 0.f && x < 0.f)) { x = -x; y = -y; }
      int bin = 0;
      bin += (0.9238795325f * y - 0.3826834324f * x >= 0.f);
      bin += (0.7071067812f * y - 0.7071067812f * x >= 0.f);
      bin += (0.3826834324f * y - 0.9238795325f * x >= 0.f);
      bin += (-x >= 0.f);
      bin += (-0.3826834324f * y - 0.9238795325f * x >= 0.f);
      bin += (-0.7071067812f * y - 0.7071067812f * x >= 0.f);
      bin += (-0.9238795325f * y - 0.3826834324f * x >= 0.f);

#pragma unroll
      for (int o = 0; o < 8; ++o) acc[o] += (bin == o) ? magn : 0.f;
    }

    // wave32 butterfly: full half-cell (128 px) sums per bin
#pragma unroll
    for (int o = 0; o < 8; ++o) {
      float v = acc[o];
      v += __shfl_xor(v, 16);
      v += __shfl_xor(v, 8);
      v += __shfl_xor(v, 4);
      v += __shfl_xor(v, 2);
      v += __shfl_xor(v, 1);
      acc[o] = v;
    }
    if (lane == 0) {
#pragma unroll
      for (int o = 0; o < 8; ++o) wavehist[wid * 8 + o] = acc[o];
    }
    __syncthreads();                            // publish wave partials

    // 4 cells x 8 bins: mean + L2-Hys, coalesced 128B store
    if (tid < 32) {
      const int cidx = tid >> 3;
      const int o    = tid & 7;
      float h = (wavehist[(2 * cidx) * 8 + o] +
                 wavehist[(2 * cidx + 1) * 8 + o]) * (1.0f / 256.0f);
      float ss = h * h;
      ss += __shfl_xor(ss, 1);
      ss += __shfl_xor(ss, 2);
      ss += __shfl_xor(ss, 4);
      float n = h / sqrtf(ss + 1e-10f);         // EPS^2 = 1e-10
      n = fminf(n, 0.2f);
      float s2 = n * n;
      s2 += __shfl_xor(s2, 1);
      s2 += __shfl_xor(s2, 2);
      s2 += __shfl_xor(s2, 4);
      const float ov = n / sqrtf(s2 + 1e-10f);
      const int cc = blk * BLK_CELLS + cidx;
      out[(((b * 32) + cr) * 32 + cc) * 8 + o] = ov;
    }
    // next iteration's leading __syncthreads (after wave0's TDM issue)
    // protects wavehist rewrite; the barrier above protects buffer reuse
  }
}

extern "C" void kernel_launch(void* const* d_in, const int* in_sizes, int n_in,
                              void* d_out, int out_size, void* d_ws, size_t ws_size,
                              hipStream_t stream) {
  (void)in_sizes; (void)n_in; (void)d_ws; (void)ws_size; (void)out_size;
  const float* fake = (const float*)d_in[0];   // [64,3,512,512] f32
  float* out = (float*)d_out;                  // [64, 32*32*8] f32
  dim3 grid(32, 64);                           // (cell_row, batch)
  dim3 block(256);
  hog_cdna5_kernel<<<grid, block, 0, stream>>>(fake, out);
}